// LLIE_loss_68496138436939
// MI455X (gfx1250) — compile-verified
//
#include <hip/hip_runtime.h>
#include <math.h>
#include <stdint.h>

#define IMG_W   512
#define HW      (512 * 512)
#define NB      16
#define NC      3
#define PLANES  (NB * NC)            // 48
#define VALID   506                  // 512 - 7 + 1
#define NTI     32                   // ceil(506/16)
#define TILES_PER_PLANE (NTI * NTI)
#define TILES_PER_PAIR  (PLANES * TILES_PER_PLANE)
#define TOTAL_TILES     (2 * TILES_PER_PAIR)   // 98304, divisible by 8

typedef __attribute__((ext_vector_type(2))) float v2f;
typedef __attribute__((ext_vector_type(8))) float v8f;

// ---------------------------------------------------------------- zero ws
__global__ void zero_ws(float* ws, int n) {
    int i = blockIdx.x * 256 + threadIdx.x;
    if (i < n) ws[i] = 0.0f;
}

// ------------------------------------------------- elementwise reductions
// ws[0] sum|comp0-real_comp0|   (N = 16*3*HW)
// ws[1] sum|img - comp0*comp1|
// ws[2] sum|target - real_comp0*real_comp1|
// ws[3] sum (comp2-real_comp1)^2 (N = 16*HW)
// ws[4] sum (img-target)^2
__global__ void elem_kernel(const float* __restrict__ img,
                            const float* __restrict__ target,
                            const float* __restrict__ comp0,
                            const float* __restrict__ comp1,
                            const float* __restrict__ comp2,
                            const float* __restrict__ real_comp0,
                            const float* __restrict__ real_comp1,
                            float* ws) {
    const long long N1 = (long long)NB * HW;
    long long p = (long long)blockIdx.x * 256 + threadIdx.x;
    float s0 = 0.f, s1 = 0.f, s2 = 0.f, s3 = 0.f, s4 = 0.f;
    for (; p < N1; p += (long long)gridDim.x * 256) {
        long long b = p / HW, sp = p % HW;
        long long base3 = b * (3LL * HW) + sp;
        float c1v  = comp1[p];
        float c2v  = comp2[p];
        float rc1v = real_comp1[p];
        float dl = c2v - rc1v;
        s3 += dl * dl;
#pragma unroll
        for (int c = 0; c < 3; ++c) {
            long long o = base3 + (long long)c * HW;
            float a  = comp0[o];
            float r  = real_comp0[o];
            float im = img[o];
            float tg = target[o];
            s0 += fabsf(a - r);
            s1 += fabsf(im - a * c1v);
            s2 += fabsf(tg - r * rc1v);
            float e = im - tg;
            s4 += e * e;
        }
    }
    for (int off = 16; off > 0; off >>= 1) {
        s0 += __shfl_xor(s0, off, 32);
        s1 += __shfl_xor(s1, off, 32);
        s2 += __shfl_xor(s2, off, 32);
        s3 += __shfl_xor(s3, off, 32);
        s4 += __shfl_xor(s4, off, 32);
    }
    if ((threadIdx.x & 31) == 0) {
        atomicAdd(&ws[0], s0);
        atomicAdd(&ws[1], s1);
        atomicAdd(&ws[2], s2);
        atomicAdd(&ws[3], s3);
        atomicAdd(&ws[4], s4);
    }
}

// ------------------------------------------------------------- histogram
// 48 planes x 16 chunks; LDS 256-bin hist per block, flushed with atomics.
__global__ void hist_kernel(const float* __restrict__ target, float* wshist) {
    __shared__ unsigned int lh[256];
    lh[threadIdx.x] = 0u;
    __syncthreads();
    const int plane = blockIdx.x >> 4;
    const int chunk = blockIdx.x & 15;
    const int CH = HW / 16;  // 16384
    const float* tp = target + (long long)plane * HW + (long long)chunk * CH;
    for (int i = threadIdx.x; i < CH; i += 256) {
        int bin = (int)rintf(tp[i] * 255.0f);   // RTE matches jnp.round
        bin = min(max(bin, 0), 255);
        atomicAdd(&lh[bin], 1u);
    }
    __syncthreads();
    atomicAdd(&wshist[plane * 256 + threadIdx.x], (float)lh[threadIdx.x]);
}

// --------------------------------------------------------------- SSIM
// One wave per 16x16 output tile. Tile fill uses the CDNA5 async
// direct-to-LDS path (GLOBAL_LOAD_ASYNC_TO_LDS_B32, ASYNCcnt) — no VGPR
// staging. Horizontal 7-tap from LDS; vertical 7-tap as a banded matmul on
// the matrix pipe: 6 chained V_WMMA_F32_16X16X4_F32 per stat map
// (5 maps -> 30 wmma per tile).
__global__ void __launch_bounds__(256)
ssim_kernel(const float* __restrict__ img,  const float* __restrict__ target,
            const float* __restrict__ comp0, const float* __restrict__ real_comp0,
            float* ws) {
    __shared__ float tilebuf[8][2][22][24];

    const int wave = threadIdx.x >> 5;
    const int lane = threadIdx.x & 31;
    const int tile = blockIdx.x * 8 + wave;   // TOTAL_TILES % 8 == 0

    const int pair = tile / TILES_PER_PAIR;
    int rem = tile % TILES_PER_PAIR;
    const int plane = rem / TILES_PER_PLANE;
    rem %= TILES_PER_PLANE;
    const int i0 = (rem / NTI) * 16;
    const int j0 = (rem % NTI) * 16;

    const float* X = pair ? img    : comp0;
    const float* Y = pair ? target : real_comp0;
    const float* xb = X + (long long)plane * HW;
    const float* yb = Y + (long long)plane * HW;

    float (*lx)[24] = tilebuf[wave][0];
    float (*ly)[24] = tilebuf[wave][1];

    // LDS byte offsets (addrspace(3) ptrtoint == LDS offset)
    const unsigned lxb = (unsigned)(uintptr_t)&lx[0][0];
    const unsigned lyb = (unsigned)(uintptr_t)&ly[0][0];

    for (int idx = lane; idx < 22 * 22; idx += 32) {
        int r = idx / 22, c = idx - r * 22;
        int gr = min(i0 + r, IMG_W - 1);
        int gc = min(j0 + c, IMG_W - 1);
        unsigned goff = (unsigned)((gr * IMG_W + gc) * 4);   // byte offset in plane
        unsigned loff = (unsigned)((r * 24 + c) * 4);        // byte offset in tile
        unsigned ldx = lxb + loff;
        unsigned ldy = lyb + loff;
        // async copy: copies 4 bytes from global memory at saddr plus the offset in vaddr to the LDS address in vdst   (GVS mode, ASYNCcnt)
        asm volatile("global_load_async_to_lds_b32 %0, %1, %2"
                     :: "v"(ldx), "v"(goff), "s"(xb) : "memory");
        asm volatile("global_load_async_to_lds_b32 %0, %1, %2"
                     :: "v"(ldy), "v"(goff), "s"(yb) : "memory");
    }
    asm volatile("s_wait_asynccnt 0x0" ::: "memory");
    __syncthreads();

    const int n    = lane & 15;          // output column / H column
    const int koff = (lane & 16) ? 2 : 0;
    const int m    = n;                  // A-matrix row for this lane

    v8f ax = {}, ay = {}, axx = {}, ayy = {}, axy = {};

#pragma unroll
    for (int kc = 0; kc < 6; ++kc) {
        float h[2][5];
#pragma unroll
        for (int s = 0; s < 2; ++s) {
            int k  = 4 * kc + koff + s;
            int kr = (k < 22) ? k : 21;  // clamped rows get A[m][k]==0 anyway
            float hx = 0.f, hy = 0.f, hxx = 0.f, hyy = 0.f, hxy = 0.f;
#pragma unroll
            for (int d = 0; d < 7; ++d) {
                float xv = lx[kr][n + d];
                float yv = ly[kr][n + d];
                hx  += xv;       hy  += yv;
                hxx += xv * xv;  hyy += yv * yv;
                hxy += xv * yv;
            }
            h[s][0] = hx; h[s][1] = hy; h[s][2] = hxx; h[s][3] = hyy; h[s][4] = hxy;
        }
        const int k0 = 4 * kc + koff;
        v2f a;
        a.x = (k0     >= m && k0     <= m + 6) ? 1.0f : 0.0f;
        a.y = (k0 + 1 >= m && k0 + 1 <= m + 6) ? 1.0f : 0.0f;
        v2f b;
        b.x = h[0][0]; b.y = h[1][0];
        ax  = __builtin_amdgcn_wmma_f32_16x16x4_f32(false, a, false, b, (short)0, ax,  false, false);
        b.x = h[0][1]; b.y = h[1][1];
        ay  = __builtin_amdgcn_wmma_f32_16x16x4_f32(false, a, false, b, (short)0, ay,  false, false);
        b.x = h[0][2]; b.y = h[1][2];
        axx = __builtin_amdgcn_wmma_f32_16x16x4_f32(false, a, false, b, (short)0, axx, false, false);
        b.x = h[0][3]; b.y = h[1][3];
        ayy = __builtin_amdgcn_wmma_f32_16x16x4_f32(false, a, false, b, (short)0, ayy, false, false);
        b.x = h[0][4]; b.y = h[1][4];
        axy = __builtin_amdgcn_wmma_f32_16x16x4_f32(false, a, false, b, (short)0, axy, false, false);
    }

    const float inv_np   = 1.0f / 49.0f;
    const float cov_norm = 49.0f / 48.0f;
    const float C1c = 0.0001f, C2c = 0.0009f;
    const int mbase = (lane & 16) ? 8 : 0;

    float ssum = 0.0f;
#pragma unroll
    for (int jj = 0; jj < 8; ++jj) {
        int io = i0 + mbase + jj;
        int jo = j0 + n;
        float ux  = ax[jj]  * inv_np;
        float uy  = ay[jj]  * inv_np;
        float uxx = axx[jj] * inv_np;
        float uyy = ayy[jj] * inv_np;
        float uxy = axy[jj] * inv_np;
        float vx  = cov_norm * (uxx - ux * ux);
        float vy  = cov_norm * (uyy - uy * uy);
        float vxy = cov_norm * (uxy - ux * uy);
        float S = ((2.f * ux * uy + C1c) * (2.f * vxy + C2c)) /
                  ((ux * ux + uy * uy + C1c) * (vx + vy + C2c));
        if (io < VALID && jo < VALID) ssum += S;
    }
    for (int off = 16; off > 0; off >>= 1) ssum += __shfl_xor(ssum, off, 32);
    if (lane == 0) atomicAdd(&ws[5 + pair], ssum);
}

// ----------------------------------------------------------- color loss
__global__ void color_kernel(const float* __restrict__ color_hist,
                             const float* __restrict__ wshist, float* ws) {
    const float invHW = 1.0f / (float)HW;
    float s = 0.0f;
    for (int i = threadIdx.x; i < PLANES * 256; i += 256)
        s += fabsf(color_hist[i] - wshist[i] * invHW);
    __shared__ float red[8];
    for (int off = 16; off > 0; off >>= 1) s += __shfl_xor(s, off, 32);
    if ((threadIdx.x & 31) == 0) red[threadIdx.x >> 5] = s;
    __syncthreads();
    if (threadIdx.x == 0) {
        float t = 0.f;
        for (int w = 0; w < 8; ++w) t += red[w];
        ws[7] = t / (float)(PLANES * 256);   // mean already
    }
}

// ------------------------------------------------------------- finalize
__global__ void final_kernel(const float* __restrict__ score,
                             const float* __restrict__ ws, float* out) {
    float g = 0.0f;
    if (threadIdx.x < NB) {
        float z = -score[threadIdx.x];
        g = fmaxf(z, 0.0f) + log1pf(expf(-fabsf(z)));   // stable softplus
    }
    for (int off = 16; off > 0; off >>= 1) g += __shfl_xor(g, off, 32);
    if (threadIdx.x == 0) {
        const float N3  = (float)NB * NC * HW;            // 12582912
        const float N1f = (float)NB * HW;                 // 4194304
        const float NS  = (float)PLANES * VALID * VALID;  // 12289728
        float gan    = g / (float)NB;
        float ssim0  = ws[5] / NS;
        float ssim1  = ws[6] / NS;
        float r_loss = (1.0f - ssim0) + ws[0] / N3 + ws[1] / N3 + ws[2] / N3;
        float light  = ws[3] / N1f;
        float color  = ws[7];
        float out_l  = (1.0f - ssim1) + ws[4] / N3;
        float loss   = 1.0f * r_loss + 1.0f * light + 0.1f * color
                     + 1.0f * out_l + 0.05f * gan;
        out[0] = loss;                 // scalar loss
        out[1] = loss;                 // breakdown[0]
        out[2] = r_loss;
        out[3] = light;
        out[4] = color;
        out[5] = out_l;
        out[6] = gan;
    }
}

// ------------------------------------------------------------------ host
extern "C" void kernel_launch(void* const* d_in, const int* in_sizes, int n_in,
                              void* d_out, int out_size, void* d_ws, size_t ws_size,
                              hipStream_t stream) {
    const float* img        = (const float*)d_in[0];
    const float* target     = (const float*)d_in[1];
    const float* comp0      = (const float*)d_in[2];
    const float* comp1      = (const float*)d_in[3];
    const float* comp2      = (const float*)d_in[4];
    const float* real_comp0 = (const float*)d_in[5];
    const float* real_comp1 = (const float*)d_in[6];
    const float* color_hist = (const float*)d_in[7];
    const float* score      = (const float*)d_in[8];

    float* ws     = (float*)d_ws;
    float* wshist = ws + 8;
    const int ws_elems = 8 + PLANES * 256;

    zero_ws<<<(ws_elems + 255) / 256, 256, 0, stream>>>(ws, ws_elems);

    elem_kernel<<<(int)(((long long)NB * HW) / 256), 256, 0, stream>>>(
        img, target, comp0, comp1, comp2, real_comp0, real_comp1, ws);

    hist_kernel<<<PLANES * 16, 256, 0, stream>>>(target, wshist);

    ssim_kernel<<<TOTAL_TILES / 8, 256, 0, stream>>>(img, target, comp0, real_comp0, ws);

    color_kernel<<<1, 256, 0, stream>>>(color_hist, wshist, ws);

    final_kernel<<<1, 32, 0, stream>>>(score, ws, (float*)d_out);
}